// QuantumGenerator_25314537242892
// MI455X (gfx1250) — compile-verified
//
#include <hip/hip_runtime.h>

// QuantumGenerator on gfx1250: each wave32 simulates one 8-qubit circuit whose
// 256-amplitude state is a 16x16 complex matrix; every RY weight layer is
// S <- sign ∘ (A · S · B^T) implemented with v_wmma_f32_16x16x4_f32 (4 chained
// K=4 WMMAs per 16x16x16 matmul, Re/Im components separately since A,B real).

#define NQ   8
#define QD   6
#define NGEN 8

typedef __attribute__((ext_vector_type(2))) float v2f;
typedef __attribute__((ext_vector_type(8))) float v8f;

__device__ __forceinline__ v8f wmma4(v2f a, v2f b, v8f c) {
    // (neg_a, A, neg_b, B, c_mod, C, reuse_a, reuse_b)
    return __builtin_amdgcn_wmma_f32_16x16x4_f32(false, a, false, b, (short)0, c,
                                                 false, false);
}

__device__ __forceinline__ float ry_elem(float t, int a, int b) {
    // RY = [[c, -s],[s, c]], c=cos(t/2), s=sin(t/2)
    float c = cosf(0.5f * t), s = sinf(0.5f * t);
    return (a == b) ? c : ((a == 1) ? s : -s);
}

// Per-(generator, layer) Kronecker factors of the RY layer:
//   A[r][c]  = prod_{i=0..3} RY(w[i])    [bit(3-i) r][bit(3-i) c]   (row wires 0-3)
//   R = B^T, B[c][c'] = prod_{i=0..3} RY(w[4+i])[bit(3-i) c][bit(3-i) c'] (col wires 4-7)
__global__ __launch_bounds__(256) void qgen_precompute(const float* __restrict__ qp,
                                                       float* __restrict__ Amat,
                                                       float* __restrict__ Rmat) {
    int gl = blockIdx.x;            // g*QD + l, 0..47
    int e  = threadIdx.x;           // 0..255
    int r  = e >> 4, c = e & 15;
    const float* w = qp + gl * NQ;
    float a = 1.f, bt = 1.f;
#pragma unroll
    for (int i = 0; i < 4; ++i) {
        int sh = 3 - i;
        int ra = (r >> sh) & 1, ca = (c >> sh) & 1;
        a  *= ry_elem(w[i],     ra, ca);
        bt *= ry_elem(w[4 + i], ca, ra);   // transposed on the fly: R = B^T
    }
    Amat[gl * 256 + e] = a;
    Rmat[gl * 256 + e] = bt;
}

__global__ __launch_bounds__(256) void qgen_main(const float* __restrict__ noise,
                                                 const float* __restrict__ Amat,
                                                 const float* __restrict__ Rmat,
                                                 float* __restrict__ out) {
    // Per-wave scratch: Sre/Sim/Tre/Tim (256 each) + vrow/vcol (32 re + 32 im)
    __shared__ float lds[8][1088];
    const int b    = blockIdx.x;          // batch sample
    const int wave = threadIdx.x >> 5;    // generator index (8 waves/block)
    const int lane = threadIdx.x & 31;
    const int g    = wave;

    float* Sre = lds[wave];
    float* Sim = Sre + 256;
    float* Tre = Sim + 256;
    float* Tim = Tre + 256;
    float* vre = Tim + 256;               // [0..15]=vrow, [16..31]=vcol
    float* vim = vre + 32;

    // ---- Noise section: product state. u_i = RX(t)RY(t)|0> = [c^2 - i s^2, sc - i sc]
    {
        const int half = lane >> 4;       // 0: wires 0-3 (rows), 1: wires 4-7 (cols)
        const int idx4 = lane & 15;
        float re = 1.f, im = 0.f;
#pragma unroll
        for (int i = 0; i < 4; ++i) {
            int bit  = (idx4 >> (3 - i)) & 1;     // wire (half*4+i) is bit (3-i)
            float t  = noise[b * NQ + half * 4 + i];
            float ch = cosf(0.5f * t), sh = sinf(0.5f * t);
            float ure = bit ? sh * ch : ch * ch;
            float uim = bit ? -sh * ch : -sh * sh;
            float nre = re * ure - im * uim;
            float nim = re * uim + im * ure;
            re = nre; im = nim;
        }
        vre[lane] = re; vim[lane] = im;
    }
    __syncthreads();

    // S0 = vrow ⊗ vcol (complex outer product), row-major in LDS
#pragma unroll
    for (int j = 0; j < 8; ++j) {
        int idx = j * 32 + lane;
        int r = idx >> 4, c = idx & 15;
        float rr = vre[r],      ri = vim[r];
        float cr = vre[16 + c], ci = vim[16 + c];
        Sre[idx] = rr * cr - ri * ci;
        Sim[idx] = rr * ci + ri * cr;
    }
    __syncthreads();

    // WMMA operand lane maps (ISA 7.12.2, 32-bit):
    //  A 16x4 : element(m,k) -> lane m + 16*(k>=2), vgpr k&1
    //  B 4x16 : element(k,n) -> lane n + 16*(k>=2), vgpr k&1
    //  D 16x16: element(m,n) -> lane n + 16*(m>=8), vgpr m&7
    const int mn    = lane & 15;          // serves as both m (A/D) and n (B/D)
    const int khalf = lane >> 4;

    for (int l = 0; l < QD; ++l) {
        const float* Ag = Amat + (g * QD + l) * 256;
        const float* Rg = Rmat + (g * QD + l) * 256;

        // ---- T = A * S (Re and Im share the A operand)
        {
            v8f aR = {}, aI = {};
#pragma unroll
            for (int kc = 0; kc < 4; ++kc) {
                int k0 = 4 * kc + 2 * khalf;
                v2f a;  a.x  = Ag[mn * 16 + k0];      a.y  = Ag[mn * 16 + k0 + 1];
                v2f br; br.x = Sre[k0 * 16 + mn];     br.y = Sre[(k0 + 1) * 16 + mn];
                v2f bi; bi.x = Sim[k0 * 16 + mn];     bi.y = Sim[(k0 + 1) * 16 + mn];
                aR = wmma4(a, br, aR);
                aI = wmma4(a, bi, aI);
            }
#pragma unroll
            for (int v = 0; v < 8; ++v) {             // D layout -> row-major LDS
                int m = v + 8 * khalf;
                Tre[m * 16 + mn] = aR[v];
                Tim[m * 16 + mn] = aI[v];
            }
        }
        __syncthreads();

        // ---- S = sign ∘ (T * R), sign(idx) = (-1)^popcount(idx & (idx>>1))  (CZ chain)
        {
            v8f aR = {}, aI = {};
#pragma unroll
            for (int kc = 0; kc < 4; ++kc) {
                int k0 = 4 * kc + 2 * khalf;
                v2f ar; ar.x = Tre[mn * 16 + k0];     ar.y = Tre[mn * 16 + k0 + 1];
                v2f ai; ai.x = Tim[mn * 16 + k0];     ai.y = Tim[mn * 16 + k0 + 1];
                v2f bb; bb.x = Rg[k0 * 16 + mn];      bb.y = Rg[(k0 + 1) * 16 + mn];
                aR = wmma4(ar, bb, aR);
                aI = wmma4(ai, bb, aI);
            }
#pragma unroll
            for (int v = 0; v < 8; ++v) {
                int m    = v + 8 * khalf;
                int idx8 = (m << 4) | mn;             // 8-bit amplitude index
                float sgn = (__popc(idx8 & (idx8 >> 1)) & 1) ? -1.f : 1.f;
                Sre[idx8] = sgn * aR[v];
                Sim[idx8] = sgn * aI[v];
            }
        }
        __syncthreads();
    }

    // ---- <X_q> = 2 * sum_pairs (Re0*Re1 + Im0*Im1), pair bit = (7-q)
#pragma unroll
    for (int q = 0; q < NQ; ++q) {
        int bp = 7 - q;
        float part = 0.f;
#pragma unroll
        for (int t4 = 0; t4 < 4; ++t4) {
            int p   = t4 * 32 + lane;                 // 0..127 pair enumeration
            int low = p & ((1 << bp) - 1);
            int hi  = p >> bp;
            int i0  = (hi << (bp + 1)) | low;
            int i1  = i0 | (1 << bp);
            part += Sre[i0] * Sre[i1] + Sim[i0] * Sim[i1];
        }
#pragma unroll
        for (int off = 16; off > 0; off >>= 1)
            part += __shfl_xor(part, off, 32);
        if (lane == 0) out[(b * NGEN + g) * NQ + q] = 2.f * part;
    }
}

extern "C" void kernel_launch(void* const* d_in, const int* in_sizes, int n_in,
                              void* d_out, int out_size, void* d_ws, size_t ws_size,
                              hipStream_t stream) {
    const float* noise = (const float*)d_in[0];   // [B, 8]
    const float* qp    = (const float*)d_in[1];   // [8, 6, 8]
    const int B = in_sizes[0] / NQ;

    float* Amat = (float*)d_ws;                   // [G*QD][256]
    float* Rmat = Amat + NGEN * QD * 256;         // [G*QD][256]  (96 KB total in d_ws)

    qgen_precompute<<<NGEN * QD, 256, 0, stream>>>(qp, Amat, Rmat);
    qgen_main<<<B, 256, 0, stream>>>(noise, Amat, Rmat, (float*)d_out);
}